// CRF_62758062129666
// MI455X (gfx1250) — compile-verified
//
#include <hip/hip_runtime.h>
#include <hip/hip_bf16.h>

// ---------------- problem constants ----------------
#define T_LEN      2000000
#define NTAGS      5
#define START_TAG  3
#define STOP_TAG   4

// ---------------- decomposition ----------------
#define CHUNK_LEN  100
#define NCHUNKS    (T_LEN / CHUNK_LEN)          // 20000 (exact)
#define CPW        3                            // chunks per wave (15 WMMA cols)
#define WAVES_P1   ((NCHUNKS + CPW - 1) / CPW)  // 6667
#define WPB        4                            // waves per block
#define BLK_P1     (WPB * 32)
#define GRID_P1    ((WAVES_P1 + WPB - 1) / WPB)

#define TS         20                           // steps per async tile
#define NT         (CHUNK_LEN / TS)             // 5 tiles per chunk
#define TILE_ELEMS (CPW * TS * NTAGS)           // 300 floats per tile
#define ISSUES     ((TILE_ELEMS + 31) / 32)     // 10 async b32 issues per tile

#define GROUP_SZ   100
#define NGROUPS    (NCHUNKS / GROUP_SZ)         // 200

#define RENORM     4                            // renormalize every 4 steps

#define GOLD_BLOCKS  1024
#define GOLD_THREADS 256

#define LN2_D 0.69314718055994530942

typedef __attribute__((ext_vector_type(2))) float v2f;
typedef __attribute__((ext_vector_type(8))) float v8f;

// Raw exponent extraction: g is always a clamped positive normal, so skip
// libm's inf/NaN fixups. frexp convention: g = m * 2^e, m in [0.5,1).
__device__ __forceinline__ int fexp2(float g) {
#if __has_builtin(__builtin_amdgcn_frexp_expf)
  return __builtin_amdgcn_frexp_expf(g);     // v_frexp_exp_i32_f32
#else
  return ilogbf(g) + 1;
#endif
}

// Async-prefetch one 3-chunk x TS-step feat tile into this wave's LDS buffer.
// GLOBAL_LOAD_ASYNC_TO_LDS_B32: VDST = LDS byte address, VADDR = 64-bit global
// address (GV mode, saddr = off). Tracked by ASYNCcnt.
__device__ __forceinline__
void issue_tile_async(const float* __restrict__ feats, float* tile,
                      int c0, int tt, int lane)
{
  #pragma unroll
  for (int j = 0; j < ISSUES; ++j) {
    int idx = lane + j * 32;
    if (idx < TILE_ELEMS) {
      int cc  = idx / (TS * NTAGS);
      int o   = idx - cc * (TS * NTAGS);
      int cid = c0 + cc; if (cid >= NCHUNKS) cid = 0;   // pad chunk: dummy read
      const float* g = feats + (size_t)cid * (CHUNK_LEN * NTAGS)
                             + tt * (TS * NTAGS) + o;
      unsigned           laddr = (unsigned)(uintptr_t)(tile + idx); // LDS offset = low 32b
      unsigned long long gaddr = (unsigned long long)(uintptr_t)g;
      asm volatile("global_load_async_to_lds_b32 %0, %1, off"
                   :: "v"(laddr), "v"(gaddr) : "memory");
    }
  }
}

// =====================================================================
// Phase 1: each wave advances 3 chunks of the scan in the scaled-exp
// domain:  S <- diag(exp(feat_t)) * exp(T) * S, via two chained
// V_WMMA_F32_16X16X4_F32 (K=5 split as K0..3 + K4..7-padded).
//   A (16x4 f32): lanes 0-15 row M=lane {v0=K0,v1=K1};
//                 lanes 16-31 row M=lane-16 {v0=K2,v1=K3}.
//   B (4x16 f32): v0: lanes0-15 = K0 row, lanes16-31 = K2 row;
//                 v1: lanes0-15 = K1 row, lanes16-31 = K3 row.
//   C/D (16x16):  vgpr m: lanes0-15 row m, lanes16-31 row m+8.
// Chunk cc owns columns 5*cc..5*cc+4; col 15 dead. Renormalization is by
// powers of two (exact), scale tracked as an integer exponent.
// =====================================================================
__global__ __launch_bounds__(BLK_P1)
void crf_phase1(const float* __restrict__ feats,
                const float* __restrict__ trans,
                float* __restrict__ mats /* NCHUNKS x 26 */)
{
  __shared__ float fbuf[WPB][2][TILE_ELEMS];

  const int lane = threadIdx.x & 31;
  const int wib  = threadIdx.x >> 5;
  const int wave = blockIdx.x * WPB + wib;
  if (wave >= WAVES_P1) return;           // wave-uniform

  const int  c0   = wave * CPW;
  const bool hi   = lane >= 16;
  const int  half = lane & 15;
  const int  cc   = half / NTAGS;         // chunk slot (col 15 -> 3, dead)
  const int  q    = half % NTAGS;
  const int  base = cc * NTAGS;
  const bool colv = (half < CPW * NTAGS) && !hi;

  // E = exp(transitions) as two 16x4 A operands (rows >=5 zero).
  // One-time setup: keep the precise expf here.
  float a1x = 0.f, a1y = 0.f, a2x = 0.f;
  if (half < NTAGS) {
    if (!hi) { a1x = expf(trans[half*NTAGS + 0]); a1y = expf(trans[half*NTAGS + 1]);
               a2x = expf(trans[half*NTAGS + 4]); }            // exp(-1e4)==0
    else     { a1x = expf(trans[half*NTAGS + 2]); a1y = expf(trans[half*NTAGS + 3]); }
  }
  v2f A1; A1.x = a1x; A1.y = a1y;
  v2f A2; A2.x = a2x; A2.y = 0.f;

  // State S (B layout), init = identity per 5-column chunk block.
  float b1x = (half < 15) && ((!hi && q == 0) || (hi && q == 2)) ? 1.f : 0.f;
  float b1y = (half < 15) && ((!hi && q == 1) || (hi && q == 3)) ? 1.f : 0.f;
  float b2x = (half < 15) && (!hi && q == 4) ? 1.f : 0.f;
  float b2y = 0.f;

  v8f d = {};
  int  iscale = 0;                        // exact power-of-two log-scale

  issue_tile_async(feats, &fbuf[wib][0][0], c0, 0, lane);

  for (int tt = 0; tt < NT; ++tt) {
    if (tt + 1 < NT) {
      issue_tile_async(feats, &fbuf[wib][(tt + 1) & 1][0], c0, tt + 1, lane);
      asm volatile("s_wait_asynccnt %0" :: "n"(ISSUES) : "memory"); // prev tile done
    } else {
      asm volatile("s_wait_asynccnt 0" ::: "memory");
    }
    const float* tile = &fbuf[wib][tt & 1][0];

    for (int ss = 0; ss < TS; ++ss) {
      const int s = tt * TS + ss;
      float f = colv ? tile[cc * (TS * NTAGS) + ss * NTAGS + q] : 0.f;
      // Fast native exp: feats ~ N(0,1), far from the |x|>88 edge cases;
      // ~2ulp/factor random-walks to <1e-3 in log space over 2M steps.
      float e = __expf(f);

      // D = E * S (two chained f32 WMMAs, K = 0..7)
      v2f B1; B1.x = b1x; B1.y = b1y;
      v2f B2; B2.x = b2x; B2.y = b2y;
      v8f z = {};
      v8f acc = __builtin_amdgcn_wmma_f32_16x16x4_f32(false, A2, false, B2,
                                                      (short)0, z,   false, false);
      d       = __builtin_amdgcn_wmma_f32_16x16x4_f32(false, A1, false, B1,
                                                      (short)0, acc, false, false);

      // Row scaling: S'[i][col] = exp(feat_{chunk(col)}[i]) * D[i][col]
      float r0 = __shfl(e, base + 0, 32);
      float r1 = __shfl(e, base + 1, 32);
      float r2 = __shfl(e, base + 2, 32);
      float r3 = __shfl(e, base + 3, 32);
      float r4 = __shfl(e, base + 4, 32);
      d[0] *= r0; d[1] *= r1; d[2] *= r2; d[3] *= r3; d[4] *= r4;

      // Exact power-of-two renorm per chunk, every RENORM steps.
      if ((s & (RENORM - 1)) == (RENORM - 1)) {
        float m = fmaxf(fmaxf(fmaxf(d[0], d[1]), fmaxf(d[2], d[3])), d[4]);
        float g = fmaxf(__shfl(m, base + 0, 32), __shfl(m, base + 1, 32));
        g = fmaxf(g, __shfl(m, base + 2, 32));
        g = fmaxf(g, __shfl(m, base + 3, 32));
        g = fmaxf(g, __shfl(m, base + 4, 32));
        g = fmaxf(g, 1e-30f);
        int ex = fexp2(g);                // v_frexp_exp_i32_f32
        float fac = ldexpf(1.f, -ex);     // exact scaling, entries -> [0.5,1)
        d[0] *= fac; d[1] *= fac; d[2] *= fac; d[3] *= fac; d[4] *= fac;
        iscale += ex;
      }

      // Repack D (rows 0..4 in lanes 0-15) -> B layout (2 half-wave moves).
      float p2 = __shfl(d[2], half, 32);
      float p3 = __shfl(d[3], half, 32);
      b1x = hi ? p2  : d[0];
      b1y = hi ? p3  : d[1];
      b2x = hi ? 0.f : d[4];
      b2y = 0.f;
    }
  }

  // Store chunk transfer matrix (exp-domain) + integer log2-scale.
  if (colv) {
    const int cid = c0 + cc;
    if (cid < NCHUNKS) {
      float* mp = mats + (size_t)cid * 26;
      #pragma unroll
      for (int i = 0; i < NTAGS; ++i) mp[i*NTAGS + q] = d[i];
      if (q == 0) mp[25] = (float)iscale;          // exact integer in float
    }
  }
}

// =====================================================================
// Phase 2: one wave folds GROUP_SZ chunk matrices (ascending t, new on
// the LEFT) into one group matrix; power-of-two renorm, integer scales.
// =====================================================================
__global__ __launch_bounds__(128)
void crf_phase2(const float* __restrict__ mats, float* __restrict__ gmats)
{
  const int lane = threadIdx.x & 31;
  const int grp  = blockIdx.x * 4 + (threadIdx.x >> 5);
  if (grp >= NGROUPS) return;

  const bool act = lane < NTAGS * NTAGS;
  const int  i   = act ? lane / NTAGS : 0;
  const int  k   = act ? lane % NTAGS : 0;

  float R  = (act && i == k) ? 1.f : 0.f;
  long  rs = 0;

  for (int c = grp * GROUP_SZ; c < (grp + 1) * GROUP_SZ; ++c) {
    const float* mp = mats + (size_t)c * 26;
    float gv = act ? mp[lane] : 0.f;
    float gs = mp[25];
    float acc = 0.f;
    #pragma unroll
    for (int j = 0; j < NTAGS; ++j) {
      float a = __shfl(gv, i * NTAGS + j, 32);
      float b = __shfl(R,  j * NTAGS + k, 32);
      acc = fmaf(a, b, acc);
    }
    acc = act ? acc : 0.f;
    float m = acc;
    #pragma unroll
    for (int off = 16; off; off >>= 1) m = fmaxf(m, __shfl_xor(m, off, 32));
    m = fmaxf(m, 1e-30f);
    int ex = fexp2(m);
    R  = acc * ldexpf(1.f, -ex);
    rs += (long)gs + ex;
  }
  if (act)       gmats[(size_t)grp * 26 + lane] = R;
  if (lane == 0) gmats[(size_t)grp * 26 + 25]   = (float)rs; // |rs| < 2^24
}

// =====================================================================
// Gold score: deterministic two-stage tree reduction, f64 accumulation.
// =====================================================================
__global__ __launch_bounds__(GOLD_THREADS)
void crf_gold(const float* __restrict__ feats, const int* __restrict__ tags,
              const float* __restrict__ trans, double* __restrict__ part)
{
  __shared__ double red[GOLD_THREADS];
  double acc = 0.0;
  const int stride = gridDim.x * blockDim.x;
  for (int t = blockIdx.x * blockDim.x + threadIdx.x; t < T_LEN; t += stride) {
    int nxt = tags[t];
    int prv = (t == 0) ? START_TAG : tags[t - 1];
    acc += (double)trans[nxt * NTAGS + prv] + (double)feats[t * NTAGS + nxt];
  }
  red[threadIdx.x] = acc;
  __syncthreads();
  for (int off = GOLD_THREADS / 2; off; off >>= 1) {
    if (threadIdx.x < off) red[threadIdx.x] += red[threadIdx.x + off];
    __syncthreads();
  }
  if (threadIdx.x == 0) part[blockIdx.x] = red[0];
}

// =====================================================================
// Final: fold 200 group matrices into exp(init)=e_START;
// alpha = log(sum_i exp(T[STOP][i]) v_i) + (total pow2 scale)*ln2 (f64).
// =====================================================================
__global__ __launch_bounds__(256)
void crf_final(const float* __restrict__ gmats, const double* __restrict__ part,
               const float* __restrict__ trans, const int* __restrict__ tags,
               float* __restrict__ out)
{
  __shared__ double red[256];
  __shared__ double goldS;

  double a = 0.0;
  for (int b = threadIdx.x; b < GOLD_BLOCKS; b += 256) a += part[b];
  red[threadIdx.x] = a;
  __syncthreads();
  for (int off = 128; off; off >>= 1) {
    if (threadIdx.x < off) red[threadIdx.x] += red[threadIdx.x + off];
    __syncthreads();
  }
  if (threadIdx.x == 0)
    goldS = red[0] + (double)trans[STOP_TAG * NTAGS + tags[T_LEN - 1]];
  __syncthreads();

  if (threadIdx.x < 32) {
    const int lane = threadIdx.x;
    float v  = (lane == START_TAG) ? 1.f : 0.f;   // exp-domain init vector
    long  sc = 0;
    const int rbase = ((lane < NTAGS) ? lane : 0) * NTAGS;
    for (int g = 0; g < NGROUPS; ++g) {
      const float* mp = gmats + (size_t)g * 26;
      float gv = (lane < NTAGS * NTAGS) ? mp[lane] : 0.f;
      float gs = mp[25];
      float nv = 0.f;
      #pragma unroll
      for (int j = 0; j < NTAGS; ++j) {
        float gj = __shfl(gv, rbase + j, 32);
        float vj = __shfl(v, j, 32);
        nv = fmaf(gj, vj, nv);
      }
      nv = (lane < NTAGS) ? nv : 0.f;
      float m = nv;
      #pragma unroll
      for (int off = 16; off; off >>= 1) m = fmaxf(m, __shfl_xor(m, off, 32));
      m = fmaxf(m, 1e-30f);
      int ex = fexp2(m);
      v   = nv * ldexpf(1.f, -ex);
      sc += (long)gs + ex;
    }
    float term = (lane < NTAGS) ? expf(trans[STOP_TAG * NTAGS + lane]) * v : 0.f;
    #pragma unroll
    for (int off = 16; off; off >>= 1) term += __shfl_xor(term, off, 32);
    if (lane == 0) {
      double alpha = log((double)term) + (double)sc * LN2_D;
      out[0] = (float)(alpha - goldS);
    }
  }
}

// =====================================================================
extern "C" void kernel_launch(void* const* d_in, const int* in_sizes, int n_in,
                              void* d_out, int out_size, void* d_ws, size_t ws_size,
                              hipStream_t stream)
{
  const float* feats = (const float*)d_in[0];
  const int*   tags  = (const int*)d_in[1];
  const float* trans = (const float*)d_in[2];
  float*       out   = (float*)d_out;

  // Workspace layout: mats (20000*26 f32) | gmats (200*26 f32) | gpart (1024 f64)
  float*  mats  = (float*)d_ws;
  float*  gmats = mats  + (size_t)NCHUNKS * 26;
  double* gpart = (double*)(gmats + (size_t)NGROUPS * 26);   // 8B-aligned (20800B)

  crf_phase1<<<GRID_P1, BLK_P1, 0, stream>>>(feats, trans, mats);
  crf_phase2<<<(NGROUPS + 3) / 4, 128, 0, stream>>>(mats, gmats);
  crf_gold  <<<GOLD_BLOCKS, GOLD_THREADS, 0, stream>>>(feats, tags, trans, gpart);
  crf_final <<<1, 256, 0, stream>>>(gmats, gpart, trans, tags, out);
}